// weight_quantize_fn_26036091748385
// MI455X (gfx1250) — compile-verified
//
#include <hip/hip_runtime.h>
#include <math.h>

// Native vector types (work with __builtin_nontemporal_* and WMMA builtins).
typedef __attribute__((ext_vector_type(2))) float v2f;
typedef __attribute__((ext_vector_type(4))) float f4;
typedef __attribute__((ext_vector_type(8))) float v8f;

#define WAVE 32
#define PART_BLOCKS 2048

// ---------------------------------------------------------------------------
// Kernel 1: grid-stride partial reduction of sum and sum-of-squares.
// b128 loads, fixed-order wave shuffle + LDS tree => deterministic.
// ---------------------------------------------------------------------------
__global__ void wq_partial_reduce(const f4* __restrict__ wv,
                                  v2f* __restrict__ part, int nv4) {
  float s = 0.f, q = 0.f;
  const int stride = gridDim.x * blockDim.x;
  for (int i = blockIdx.x * blockDim.x + threadIdx.x; i < nv4; i += stride) {
    f4 v = wv[i];
    s += (v.x + v.y) + (v.z + v.w);
    q = fmaf(v.x, v.x, q);
    q = fmaf(v.y, v.y, q);
    q = fmaf(v.z, v.z, q);
    q = fmaf(v.w, v.w, q);
  }
#pragma unroll
  for (int off = 16; off > 0; off >>= 1) {
    s += __shfl_xor(s, off, WAVE);
    q += __shfl_xor(q, off, WAVE);
  }
  __shared__ float ss[8], qs[8];  // 256 threads = 8 waves (wave32)
  const int wid = threadIdx.x >> 5;
  if ((threadIdx.x & 31) == 0) { ss[wid] = s; qs[wid] = q; }
  __syncthreads();
  if (threadIdx.x == 0) {
    float S = 0.f, Q = 0.f;
#pragma unroll
    for (int i = 0; i < 8; ++i) { S += ss[i]; Q += qs[i]; }
    v2f p; p.x = S; p.y = Q;
    part[blockIdx.x] = p;
  }
}

// ---------------------------------------------------------------------------
// Kernel 2: one wave. Each lane accumulates PART_BLOCKS/32 partials, then the
// 32 lane values are reduced with V_WMMA_F32_16X16X4_F32:
//   A[m,0] = p_m (lanes 0..15, VGPR0), A[m,2] = p_{m+16}; A[m,1]=A[m,3]=0.
//   B = ones(4x16)  =>  D[m,n] = p_m + p_{m+16}  (exact f32 FMA math).
// D layout: VGPR r holds M=r (lanes 0..15) / M=r+8 (lanes 16..31), so
// sum(acc[0..7]) + shfl_xor(...,16) gives the full 32-lane sum on every lane.
// ---------------------------------------------------------------------------
__global__ void wq_finalize(const v2f* __restrict__ part, int nPart,
                            const float* __restrict__ alpha_p,
                            float* __restrict__ stats, float n) {
  const int lane = threadIdx.x;  // 32 threads, full wave, EXEC all ones
  float s = 0.f, q = 0.f;
  for (int i = lane; i < nPart; i += WAVE) {  // nPart % 32 == 0 -> no divergence
    v2f p = part[i];
    s += p.x;
    q += p.y;
  }
  v2f as; as.x = s; as.y = 0.f;
  v2f aq; aq.x = q; aq.y = 0.f;
  v2f ones; ones.x = 1.f; ones.y = 1.f;
  v8f cs = {};
  v8f cq = {};
  cs = __builtin_amdgcn_wmma_f32_16x16x4_f32(false, as, false, ones,
                                             (short)0, cs, false, false);
  cq = __builtin_amdgcn_wmma_f32_16x16x4_f32(false, aq, false, ones,
                                             (short)0, cq, false, false);
  float ts = ((cs[0] + cs[1]) + (cs[2] + cs[3])) +
             ((cs[4] + cs[5]) + (cs[6] + cs[7]));
  float tq = ((cq[0] + cq[1]) + (cq[2] + cq[3])) +
             ((cq[4] + cq[5]) + (cq[6] + cq[7]));
  ts += __shfl_xor(ts, 16, WAVE);
  tq += __shfl_xor(tq, 16, WAVE);
  if (lane == 0) {
    float mean = ts / n;
    float var = (tq - ts * ts / n) / (n - 1.0f);  // unbiased (ddof=1)
    float stdv = sqrtf(var);
    float alpha = alpha_p[0];
    stats[0] = mean;
    stats[1] = 1.0f / (stdv * alpha);   // normalize scale
    stats[2] = alpha * (1.0f / 7.0f);   // output scale = alpha * SF
  }
}

// ---------------------------------------------------------------------------
// Kernel 3: one thread per 16-element group. Quantize + stable-sort-equivalent
// term-budget cascade entirely in registers. NT loads (last use of weight,
// keep L2 clean) and NT stores (output never re-read, don't evict weight).
// ---------------------------------------------------------------------------
__global__ void wq_tr_quantize(const float* __restrict__ w,
                               float* __restrict__ out,
                               const float* __restrict__ stats,
                               const int* __restrict__ nt_p, int nGroups) {
  const int g = blockIdx.x * blockDim.x + threadIdx.x;
  if (g >= nGroups) return;
  const float mean = stats[0];
  const float rcp = stats[1];
  const float oscale = stats[2];
  const int budget = nt_p[0] * 16;  // num_terms * GROUP

  const f4* wv = (const f4*)w;
  f4* ov = (f4*)out;
  const int base = g * 4;  // 4 float4 per group

  f4 v[4];
#pragma unroll
  for (int j = 0; j < 4; ++j) v[j] = __builtin_nontemporal_load(&wv[base + j]);

  float ic[16];
  int q[16];
#pragma unroll
  for (int i = 0; i < 16; ++i) {
    float x = v[i >> 2][i & 3];
    float t = (x - mean) * rcp;          // ((w-mean)/std)/alpha
    t = fminf(fmaxf(t, -1.0f), 1.0f);    // clip
    ic[i] = t;
    q[i] = (int)rintf(fabsf(t) * 7.0f);  // jnp.round == round-half-even
  }

  int n4 = 0, n2 = 0;
#pragma unroll
  for (int i = 0; i < 16; ++i) {
    n4 += (q[i] >> 2) & 1;
    n2 += (q[i] >> 1) & 1;
  }

  // Stable descending sort of terms {4,2,1} == rank cascade:
  //   4-term rank = prefix4;  2-term rank = n4 + prefix2;  1-term rank = n4+n2+prefix1
  int p4 = 0, p2 = 0, p1 = 0;
  f4 o[4];
#pragma unroll
  for (int i = 0; i < 16; ++i) {
    const int b4 = (q[i] >> 2) & 1;
    const int b2 = (q[i] >> 1) & 1;
    const int b1 = q[i] & 1;
    const int k4 = (b4 && (p4 < budget)) ? 4 : 0;
    const int k2 = (b2 && (n4 + p2 < budget)) ? 2 : 0;
    const int k1 = (b1 && (n4 + n2 + p1 < budget)) ? 1 : 0;
    p4 += b4;
    p2 += b2;
    p1 += b1;
    // out = qk * SF * sign(ic) * alpha  (copysign reproduces 0*-1 = -0.0 too)
    o[i >> 2][i & 3] = copysignf((float)(k4 + k2 + k1) * oscale, ic[i]);
  }

#pragma unroll
  for (int j = 0; j < 4; ++j)
    __builtin_nontemporal_store(o[j], &ov[base + j]);
}

// ---------------------------------------------------------------------------
extern "C" void kernel_launch(void* const* d_in, const int* in_sizes, int n_in,
                              void* d_out, int out_size, void* d_ws,
                              size_t ws_size, hipStream_t stream) {
  const float* w = (const float*)d_in[0];      // weight (f32)
  const float* alpha = (const float*)d_in[1];  // wgt_alpha (f32 scalar)
  const int* nt = (const int*)d_in[2];         // num_terms (int scalar)
  float* out = (float*)d_out;

  const int n = in_sizes[0];  // 4096*8192
  const int nv4 = n / 4;

  v2f* part = (v2f*)d_ws;                                        // 2048 * 8B
  float* stats = (float*)((char*)d_ws + PART_BLOCKS * sizeof(v2f));  // 3 floats

  wq_partial_reduce<<<PART_BLOCKS, 256, 0, stream>>>((const f4*)w, part, nv4);
  wq_finalize<<<1, WAVE, 0, stream>>>(part, PART_BLOCKS, alpha, stats, (float)n);

  const int nGroups = n / 16;
  wq_tr_quantize<<<(nGroups + 255) / 256, 256, 0, stream>>>(w, out, stats, nt,
                                                            nGroups);
}